// ChamferDistance_60662118088777
// MI455X (gfx1250) — compile-verified
//
#include <hip/hip_runtime.h>

typedef __attribute__((ext_vector_type(2))) float v2f;
typedef __attribute__((ext_vector_type(8))) float v8f;

// Single-instruction float mins (skip IEEE sNaN canonicalization: WMMA outputs
// here are always quiet / finite, so minnum semantics are preserved).
__device__ __forceinline__ float fmin2_raw(float a, float b) {
    float r;
    asm("v_min_num_f32 %0, %1, %2" : "=v"(r) : "v"(a), "v"(b));
    return r;
}
__device__ __forceinline__ float fmin3_raw(float a, float b, float c) {
    float r;
    asm("v_min3_num_f32 %0, %1, %2, %3" : "=v"(r) : "v"(a), "v"(b), "v"(c));
    return r;
}

// Each 32-thread block (one wave) owns 64 rows (4 x 16-row tiles) of X for one
// batch, sweeps all 16-col tiles of Y two at a time, and computes per-row
// min_j |x_i - y_j|^2 with V_WMMA_F32_16X16X4_F32:
//   A row i = [-2x0, -2x1, -2x2, 0]   (16x4)
//   B col j = [ y0,  y1,  y2, 0]      (4x16)
//   C       = |y_j|^2 splat (column-constant == lane-constant)
//   D_ij    = |y_j|^2 - 2 x.y  -> min3-fold over j-tiles, + |x_i|^2 at the end
__global__ __launch_bounds__(32) void chamfer_rowmin_wmma(
    const float* __restrict__ X,   // [B, Nrows, 3]
    const float* __restrict__ Y,   // [B, Mcols, 3]
    float* __restrict__ outmin,    // [B, Nrows]
    int Nrows, int Mcols)
{
    const int strips = Nrows >> 6;              // 64 rows per wave
    const int b      = blockIdx.x / strips;
    const int base   = (blockIdx.x % strips) << 6;

    const int lane = threadIdx.x;               // wave32
    const int half = lane >> 4;                 // 0: K=0,1 ; 1: K=2,3
    const int n16  = lane & 15;

    // ---- A operands for 4 row tiles (ISA 7.12.2, 32-bit A 16x4 layout)
    v2f a[4];
#pragma unroll
    for (int t = 0; t < 4; ++t) {
        const int row = base + t * 16 + n16;
        const size_t xi = ((size_t)b * Nrows + row) * 3;
        const float x0 = X[xi + 0], x1 = X[xi + 1], x2 = X[xi + 2];
        a[t].x = half ? (-2.0f * x2) : (-2.0f * x0);
        a[t].y = half ? 0.0f         : (-2.0f * x1);
    }

    const float INF = __builtin_inff();
    v8f rmin[4];
#pragma unroll
    for (int t = 0; t < 4; ++t)
#pragma unroll
        for (int k = 0; k < 8; ++k) rmin[t][k] = INF;

    // ---- Sweep column tiles of Y, two tiles per iteration
    const int ntiles = Mcols >> 4;              // multiple of 2
    for (int jt = 0; jt < ntiles; jt += 2) {
        const size_t ybase = (size_t)b * Mcols;

        const size_t yi0 = (ybase + (size_t)(jt << 4) + n16) * 3;
        const float p0 = Y[yi0 + 0], p1 = Y[yi0 + 1], p2 = Y[yi0 + 2];
        const size_t yi1 = yi0 + 48;            // next 16-col tile (16*3 floats)
        const float q0 = Y[yi1 + 0], q1 = Y[yi1 + 1], q2 = Y[yi1 + 2];

        const float psq = p0 * p0 + p1 * p1 + p2 * p2;
        const float qsq = q0 * q0 + q1 * q1 + q2 * q2;

        v2f bv0, bv1;                            // B 4x16, same K half-split as A
        bv0.x = half ? p2   : p0;
        bv0.y = half ? 0.0f : p1;
        bv1.x = half ? q2   : q0;
        bv1.y = half ? 0.0f : q1;

        const v8f c0 = {psq, psq, psq, psq, psq, psq, psq, psq};
        const v8f c1 = {qsq, qsq, qsq, qsq, qsq, qsq, qsq, qsq};

#pragma unroll
        for (int t = 0; t < 4; ++t) {
            v8f d0 = __builtin_amdgcn_wmma_f32_16x16x4_f32(
                false, a[t], false, bv0, (short)0, c0, false, false);
            v8f d1 = __builtin_amdgcn_wmma_f32_16x16x4_f32(
                false, a[t], false, bv1, (short)0, c1, false, false);
#pragma unroll
            for (int k = 0; k < 8; ++k)
                rmin[t][k] = fmin3_raw(rmin[t][k], d0[k], d1[k]);
        }
    }

    // ---- Cross-lane row reduction within each 16-lane half, then + |x_i|^2
#pragma unroll
    for (int t = 0; t < 4; ++t) {
#pragma unroll
        for (int k = 0; k < 8; ++k) {
            float v = rmin[t][k];
            v = fmin2_raw(v, __shfl_xor(v, 1, 32));
            v = fmin2_raw(v, __shfl_xor(v, 2, 32));
            v = fmin2_raw(v, __shfl_xor(v, 4, 32));
            v = fmin2_raw(v, __shfl_xor(v, 8, 32));
            if (n16 == 0) {                      // lane 0 -> rows M=k, lane 16 -> rows M=k+8
                const int row = base + t * 16 + half * 8 + k;
                const size_t xi = ((size_t)b * Nrows + row) * 3;
                const float x0 = X[xi + 0], x1 = X[xi + 1], x2 = X[xi + 2];
                outmin[(size_t)b * Nrows + row] = v + (x0 * x0 + x1 * x1 + x2 * x2);
            }
        }
    }
}

// Deterministic final reduction: scalar = mean(ws1) + mean(ws2)
__global__ __launch_bounds__(256) void chamfer_reduce(
    const float* __restrict__ ws1, int n1,
    const float* __restrict__ ws2, int n2,
    float* __restrict__ out)
{
    __shared__ float s1[256];
    __shared__ float s2[256];
    const int tid = threadIdx.x;
    float acc1 = 0.0f, acc2 = 0.0f;
    for (int i = tid; i < n1; i += 256) acc1 += ws1[i];
    for (int i = tid; i < n2; i += 256) acc2 += ws2[i];
    s1[tid] = acc1; s2[tid] = acc2;
    __syncthreads();
    for (int off = 128; off > 0; off >>= 1) {
        if (tid < off) { s1[tid] += s1[tid + off]; s2[tid] += s2[tid + off]; }
        __syncthreads();
    }
    if (tid == 0)
        out[0] = s1[0] / (float)n1 + s2[0] / (float)n2;
}

extern "C" void kernel_launch(void* const* d_in, const int* in_sizes, int n_in,
                              void* d_out, int out_size, void* d_ws, size_t ws_size,
                              hipStream_t stream) {
    const float* xyz1 = (const float*)d_in[0];   // [B, N, 3] fp32
    const float* xyz2 = (const float*)d_in[1];   // [B, M, 3] fp32
    float* out = (float*)d_out;                  // scalar fp32

    const int B = 4;
    const int N = in_sizes[0] / (3 * B);         // 8192
    const int M = in_sizes[1] / (3 * B);         // 8192

    float* ws1 = (float*)d_ws;                   // [B*N] row mins (dist1)
    float* ws2 = ws1 + (size_t)B * N;            // [B*M] col mins (dist2)

    // dist1: for each x_i, min over y
    chamfer_rowmin_wmma<<<dim3(B * (N / 64)), dim3(32), 0, stream>>>(
        xyz1, xyz2, ws1, N, M);
    // dist2: same kernel with roles swapped
    chamfer_rowmin_wmma<<<dim3(B * (M / 64)), dim3(32), 0, stream>>>(
        xyz2, xyz1, ws2, M, N);

    chamfer_reduce<<<dim3(1), dim3(256), 0, stream>>>(
        ws1, B * N, ws2, B * M, out);
}